// CategNet_4312147165694
// MI455X (gfx1250) — compile-verified
//
#include <hip/hip_runtime.h>

// N = 16777216 = 2^24 ; CATEGORY_NUM = 100000 ; EPS = 1e-10
#define N_TOTAL   16777216
#define EPS_F     1e-10f
#define ELEMS_PER_THREAD 8
#define BLOCK     256
#define NBLOCKS   (N_TOTAL / (BLOCK * ELEMS_PER_THREAD))   // 8192

typedef __attribute__((ext_vector_type(2))) float v2f;
typedef __attribute__((ext_vector_type(4))) float v4f;
typedef __attribute__((ext_vector_type(8))) float v8f;

// ---------------------------------------------------------------------------
// Wave32 sum reduction. WMMA path: A(16x4) with VGPR0 = partial, VGPR1 = 0;
// B = all-ones (layout-agnostic). D[m][n] = partial[lane m] + partial[lane m+16].
// Lane's 8 D regs cover rows {0..7} (lanes 0-15) or {8..15} (lanes 16-31);
// summing them + shfl_xor(16) gives the full 32-lane sum on every lane.
// ---------------------------------------------------------------------------
__device__ __forceinline__ float wave_sum(float v) {
#if __has_builtin(__builtin_amdgcn_wmma_f32_16x16x4_f32)
  v2f a; a[0] = v;    a[1] = 0.0f;
  v2f b; b[0] = 1.0f; b[1] = 1.0f;
  v8f c = {};
  v8f d = __builtin_amdgcn_wmma_f32_16x16x4_f32(
      /*neg_a=*/false, a, /*neg_b=*/false, b,
      /*c_mod=*/(short)0, c, /*reuse_a=*/false, /*reuse_b=*/false);
  float s = ((d[0] + d[1]) + (d[2] + d[3])) + ((d[4] + d[5]) + (d[6] + d[7]));
  s += __shfl_xor(s, 16, 32);
  return s;
#else
  #pragma unroll
  for (int off = 16; off > 0; off >>= 1) v += __shfl_xor(v, off, 32);
  return v;
#endif
}

// ---------------------------------------------------------------------------
// Kernel A: zero the f64 accumulators (ws is poisoned, not re-zeroed by harness)
// ---------------------------------------------------------------------------
__global__ void categ_init(double* __restrict__ acc) {
  acc[0] = 0.0;
  acc[1] = 0.0;
}

// ---------------------------------------------------------------------------
// Kernel B: sum / sum-of-squares of bias[idx[i]] over all N.
// Regular-temporal idx loads intentionally fill L2 (64MB fits in 192MB) so the
// second pass hits L2. Gather table (400KB) stays cache-hot.
// ---------------------------------------------------------------------------
__global__ __launch_bounds__(BLOCK) void categ_reduce(
    const int*   __restrict__ idx,
    const float* __restrict__ bias,
    double*      __restrict__ acc) {
  const int tid  = blockIdx.x * BLOCK + threadIdx.x;
  const int4* idx4 = (const int4*)idx;

  int4 i0 = idx4[tid * 2 + 0];
  int4 i1 = idx4[tid * 2 + 1];

  float v0 = bias[i0.x], v1 = bias[i0.y], v2 = bias[i0.z], v3 = bias[i0.w];
  float v4 = bias[i1.x], v5 = bias[i1.y], v6 = bias[i1.z], v7 = bias[i1.w];

  float s = ((v0 + v1) + (v2 + v3)) + ((v4 + v5) + (v6 + v7));
  float q = ((v0 * v0 + v1 * v1) + (v2 * v2 + v3 * v3)) +
            ((v4 * v4 + v5 * v5) + (v6 * v6 + v7 * v7));

  // full wave active here -> EXEC all ones, legal for WMMA
  float ws = wave_sum(s);
  float wq = wave_sum(q);

  __shared__ float redS[BLOCK / 32];
  __shared__ float redQ[BLOCK / 32];
  const int lane = threadIdx.x & 31;
  const int wave = threadIdx.x >> 5;
  if (lane == 0) { redS[wave] = ws; redQ[wave] = wq; }
  __syncthreads();
  if (threadIdx.x == 0) {
    double S = 0.0, Q = 0.0;
    #pragma unroll
    for (int w = 0; w < BLOCK / 32; ++w) { S += (double)redS[w]; Q += (double)redQ[w]; }
    atomicAdd(&acc[0], S);   // global_atomic_add_f64
    atomicAdd(&acc[1], Q);
  }
}

// ---------------------------------------------------------------------------
// Kernel C: scalars mean and 1/max(sqrt(var), EPS)
// ---------------------------------------------------------------------------
__global__ void categ_stats(const double* __restrict__ acc,
                            float* __restrict__ params) {
  const double inv_n = 1.0 / (double)N_TOTAL;
  double mean = acc[0] * inv_n;
  double var  = acc[1] * inv_n - mean * mean;
  if (var < 0.0) var = 0.0;
  float stdv = (float)sqrt(var);
  if (stdv < EPS_F) stdv = EPS_F;
  params[0] = (float)mean;
  params[1] = 1.0f / stdv;
}

// ---------------------------------------------------------------------------
// Kernel D: out[i] = (bias[idx[i]] - mean) * rstd.
// idx reads hit L2 (loaded in pass B); output stored non-temporally so the
// 64MB stream doesn't evict idx/bias from L2 mid-pass.
// ---------------------------------------------------------------------------
__global__ __launch_bounds__(BLOCK) void categ_norm(
    const int*   __restrict__ idx,
    const float* __restrict__ bias,
    const float* __restrict__ params,
    float*       __restrict__ out) {
  const float mean = params[0];
  const float rstd = params[1];
  const int tid = blockIdx.x * BLOCK + threadIdx.x;
  const int4* idx4 = (const int4*)idx;

  int4 i0 = idx4[tid * 2 + 0];
  int4 i1 = idx4[tid * 2 + 1];

  v4f o0, o1;
  o0[0] = (bias[i0.x] - mean) * rstd;
  o0[1] = (bias[i0.y] - mean) * rstd;
  o0[2] = (bias[i0.z] - mean) * rstd;
  o0[3] = (bias[i0.w] - mean) * rstd;
  o1[0] = (bias[i1.x] - mean) * rstd;
  o1[1] = (bias[i1.y] - mean) * rstd;
  o1[2] = (bias[i1.z] - mean) * rstd;
  o1[3] = (bias[i1.w] - mean) * rstd;

  v4f* out4 = (v4f*)out;
  __builtin_nontemporal_store(o0, &out4[tid * 2 + 0]);
  __builtin_nontemporal_store(o1, &out4[tid * 2 + 1]);
}

// ---------------------------------------------------------------------------
extern "C" void kernel_launch(void* const* d_in, const int* in_sizes, int n_in,
                              void* d_out, int out_size, void* d_ws, size_t ws_size,
                              hipStream_t stream) {
  const int*   idx  = (const int*)d_in[0];     // [N,1] int32
  const float* bias = (const float*)d_in[1];   // [100000,1] f32
  float* out = (float*)d_out;                  // [N,1] f32

  double* acc    = (double*)d_ws;                      // 2 x f64
  float*  params = (float*)((char*)d_ws + 16);         // mean, rstd

  categ_init  <<<1,       1,     0, stream>>>(acc);
  categ_reduce<<<NBLOCKS, BLOCK, 0, stream>>>(idx, bias, acc);
  categ_stats <<<1,       1,     0, stream>>>(acc, params);
  categ_norm  <<<NBLOCKS, BLOCK, 0, stream>>>(idx, bias, params, out);
}